// ACSL_18872086298737
// MI455X (gfx1250) — compile-verified
//
#include <hip/hip_runtime.h>
#include <hip/hip_bf16.h>

// ACSL loss for MI455X (gfx1250, wave32).
// Memory-bound single pass over 16384x1204 f32 logits (79 MB @ 23.3 TB/s ~ 3.4us).
// Coalesced float4 non-temporal loads, branchless per-element math with raw
// v_exp_f32/v_log_f32 transcendentals, per-lane f32 accumulation, wave reduction
// via V_WMMA_F32_16X16X4_F32 (A = partials, B = ones => D[m][n] = s_m + s_{m+16}),
// cross-half combine via ds_swizzle SWAPX16, then deterministic fixed-order
// block/grid combine (no float atomics).

#define C_CLASSES 1204
#define N_ROWS    16384
#define VEC4      (C_CLASSES / 4)      // 301 float4 chunks per row
#define NBLOCKS   1024
#define NTHREADS  256
#define WAVES_PER_BLOCK (NTHREADS / 32)

#define LOG2E_F 1.4426950408889634f
#define LN2_F   0.6931471805599453f

typedef float v2f __attribute__((ext_vector_type(2)));
typedef float v8f __attribute__((ext_vector_type(8)));
typedef float f4  __attribute__((ext_vector_type(4)));

__device__ __forceinline__ float bits_as_float(int i) {
    union { int i; float f; } u; u.i = i; return u.f;
}
__device__ __forceinline__ int float_as_bits(float f) {
    union { int i; float f; } u; u.f = f; return u.i;
}

// Wave32 sum-reduction using the f32 WMMA unit.
// A-matrix 16x4 layout: lane m (0..15) holds A[m][0] in v0, A[m][1] in v1;
// lane m+16 holds A[m][2], A[m][3]. Put partial in v0, zero in v1, B = ones:
//   D[m][n] = A[m][0] + A[m][2] = s_m + s_{m+16}   (broadcast over n)
// Summing the 8 C/D VGPRs gives lanes 0-15 : sum(s_0..7 + s_16..23),
// lanes 16-31 : sum(s_8..15 + s_24..31); one SWAPX16 ds_swizzle completes it.
__device__ __forceinline__ float wave_reduce_wmma(float s) {
    v2f a; a.x = s;    a.y = 0.0f;
    v2f b; b.x = 1.0f; b.y = 1.0f;
    v8f c = {};
    v8f d = __builtin_amdgcn_wmma_f32_16x16x4_f32(
        /*neg_a=*/false, a, /*neg_b=*/false, b,
        /*c_mod=*/(short)0, c, /*reuse_a=*/false, /*reuse_b=*/false);
    float t = ((d[0] + d[1]) + (d[2] + d[3])) + ((d[4] + d[5]) + (d[6] + d[7]));
    // group-of-32 swizzle, xor_mask=0x10, and_mask=0x1f -> swap lane halves
    t += bits_as_float(__builtin_amdgcn_ds_swizzle(float_as_bits(t), 0x401F));
    return t;
}

// stable softplus via hardware transcendentals:
//   logaddexp(0,x) = max(x,0) + ln2 * log2(1 + 2^(-|x|*log2e))
// t in (0,1]; precision loss of log(1+t) for tiny t is absolutely negligible
// against the max(x,0) = |x| >= 16 term it accompanies.
__device__ __forceinline__ float softplus_fast(float x) {
    const float t = __builtin_amdgcn_exp2f(-fabsf(x) * LOG2E_F);
    return fmaxf(x, 0.0f) + LN2_F * __builtin_amdgcn_logf(1.0f + t);
}

__global__ __launch_bounds__(NTHREADS)
void acsl_partial_kernel(const float* __restrict__ logits,
                         const int* __restrict__ labels,
                         const unsigned char* __restrict__ rare_sel,
                         const unsigned char* __restrict__ common_sel,
                         const int* __restrict__ rare_vec,
                         const int* __restrict__ common_vec,
                         const int* __restrict__ freq_vec,
                         float* __restrict__ partials) {
    // Packed per-column bg-weight bits, 4 columns per dword:
    // bit0 = freq, bit1 = rare, bit2 = common, all at permuted index j=(c+1)%C.
    // Original col C-1 (permuted col 0) is forced to weight 1 -> bits = 1.
    __shared__ unsigned wpack[VEC4];
    __shared__ float    wsum[WAVES_PER_BLOCK];

    const int tid = threadIdx.x;
    for (int v = tid; v < VEC4; v += NTHREADS) {
        unsigned p = 0;
#pragma unroll
        for (int k = 0; k < 4; ++k) {
            const int c = 4 * v + k;
            const int j = c + 1;                 // permuted column (wraps only at c=C-1)
            unsigned bits;
            if (j == C_CLASSES) {
                bits = 1u;                        // permuted col 0: bg weight forced to 1
            } else {
                bits = (freq_vec[j]   != 0 ? 1u : 0u)
                     | (rare_vec[j]   != 0 ? 2u : 0u)
                     | (common_vec[j] != 0 ? 4u : 0u);
            }
            p |= bits << (8 * k);
        }
        wpack[v] = p;
    }
    __syncthreads();

    const int lane   = tid & 31;
    const int wave   = tid >> 5;
    const int gwave  = blockIdx.x * WAVES_PER_BLOCK + wave;
    const int nwaves = gridDim.x * WAVES_PER_BLOCK;
    const float HS_THR = 0.8472978603872034f;    // logit(0.7)

    float acc = 0.0f;
    for (int row = gwave; row < N_ROWS; row += nwaves) {
        const int   lab  = labels[row];
        const bool  isbg = (lab == C_CLASSES - 1);
        const unsigned rowmask = 1u
                               | (rare_sel[row]   ? 2u : 0u)
                               | (common_sel[row] ? 4u : 0u);
        const f4* rp = (const f4*)(logits + (size_t)row * C_CLASSES);

        for (int v = lane; v < VEC4; v += 32) {
            const f4 x4 = __builtin_nontemporal_load(rp + v);  // read-once stream
            const unsigned wp = wpack[v];
#pragma unroll
            for (int k = 0; k < 4; ++k) {
                const int   c  = 4 * v + k;
                const float x  = x4[k];
                const float oh  = (lab == c) ? 1.0f : 0.0f;
                const float bce = softplus_fast(x) - x * oh;
                // branchless weight select: both sides are a couple of VALU ops,
                // ternary lowers to v_cmp + v_cndmask (no EXEC divergence).
                const float wbg = ((wp >> (8 * k)) & rowmask) ? 1.0f : 0.0f;
                const float wfg = fmaxf((x >= HS_THR) ? 1.0f : 0.0f, oh);
                const float w   = isbg ? wbg : wfg;
                acc = fmaf(w, bce, acc);
            }
        }
    }

    const float wtot = wave_reduce_wmma(acc);    // uniform, EXEC all-ones here
    if (lane == 0) wsum[wave] = wtot;
    __syncthreads();
    if (tid == 0) {
        float s = 0.0f;
#pragma unroll
        for (int i = 0; i < WAVES_PER_BLOCK; ++i) s += wsum[i];
        partials[blockIdx.x] = s;
    }
}

__global__ __launch_bounds__(NTHREADS)
void acsl_final_kernel(const float* __restrict__ partials, int n,
                       float* __restrict__ out) {
    __shared__ float wsum[WAVES_PER_BLOCK];
    const int tid  = threadIdx.x;
    const int lane = tid & 31;
    const int wave = tid >> 5;

    float acc = 0.0f;
    for (int i = tid; i < n; i += NTHREADS) acc += partials[i];

    const float wtot = wave_reduce_wmma(acc);
    if (lane == 0) wsum[wave] = wtot;
    __syncthreads();
    if (tid == 0) {
        float s = 0.0f;
#pragma unroll
        for (int i = 0; i < WAVES_PER_BLOCK; ++i) s += wsum[i];
        out[0] = s * (1.0f / (float)N_ROWS);
    }
}

extern "C" void kernel_launch(void* const* d_in, const int* in_sizes, int n_in,
                              void* d_out, int out_size, void* d_ws, size_t ws_size,
                              hipStream_t stream) {
    const float*         logits     = (const float*)d_in[0];
    const int*           labels     = (const int*)d_in[1];           // int32 on device
    const unsigned char* rare_sel   = (const unsigned char*)d_in[2]; // 1-byte bools
    const unsigned char* common_sel = (const unsigned char*)d_in[3];
    const int*           rare_vec   = (const int*)d_in[4];
    const int*           common_vec = (const int*)d_in[5];
    const int*           freq_vec   = (const int*)d_in[6];
    float* out      = (float*)d_out;
    float* partials = (float*)d_ws;   // NBLOCKS floats of scratch

    acsl_partial_kernel<<<NBLOCKS, NTHREADS, 0, stream>>>(
        logits, labels, rare_sel, common_sel, rare_vec, common_vec, freq_vec, partials);
    acsl_final_kernel<<<1, NTHREADS, 0, stream>>>(partials, NBLOCKS, out);
}